// GraphEncoder_14439680049660
// MI455X (gfx1250) — compile-verified
//
#include <hip/hip_runtime.h>

#define IN_DIM 256
#define HID    128
#define NREL   3

typedef float v2f __attribute__((ext_vector_type(2)));
typedef float v8f __attribute__((ext_vector_type(8)));

// ---------------------------------------------------------------- utilities
__device__ __forceinline__ float wave_sum32(float v) {
#pragma unroll
  for (int off = 16; off > 0; off >>= 1) v += __shfl_xor(v, off, 32);
  return v;
}

// ------------------------------------------------- edge counts (per dst,rel)
__global__ void count_kernel(const long long* __restrict__ ei,
                             const long long* __restrict__ et,
                             unsigned* __restrict__ cnt, int E) {
  int e = blockIdx.x * blockDim.x + threadIdx.x;
  if (e >= E) return;
  int dst = (int)ei[(size_t)E + e];
  int r   = (int)et[e];
  atomicAdd(&cnt[(size_t)dst * NREL + r], 1u);
}

// ----------------------------------------------- prefix scan (CSR offsets)
// Pass 1: per-256-block totals
__global__ void scan_partials(const unsigned* __restrict__ cnt,
                              unsigned* __restrict__ blocksum, int S) {
  __shared__ unsigned sh[256];
  int i = blockIdx.x * 256 + threadIdx.x;
  sh[threadIdx.x] = (i < S) ? cnt[i] : 0u;
  __syncthreads();
  for (int off = 128; off > 0; off >>= 1) {
    if (threadIdx.x < off) sh[threadIdx.x] += sh[threadIdx.x + off];
    __syncthreads();
  }
  if (threadIdx.x == 0) blocksum[blockIdx.x] = sh[0];
}

// Pass 2: single-block exclusive scan of block totals (nb ~ 1200)
__global__ void scan_blocksums(unsigned* __restrict__ bs, int nb,
                               unsigned* __restrict__ seg_off, int S) {
  __shared__ unsigned sh[1024];
  unsigned carry = 0;
  for (int base = 0; base < nb; base += 1024) {
    int i = base + (int)threadIdx.x;
    unsigned v = (i < nb) ? bs[i] : 0u;
    sh[threadIdx.x] = v;
    __syncthreads();
    for (int off = 1; off < 1024; off <<= 1) {
      unsigned t = (threadIdx.x >= (unsigned)off) ? sh[threadIdx.x - off] : 0u;
      __syncthreads();
      sh[threadIdx.x] += t;
      __syncthreads();
    }
    if (i < nb) bs[i] = carry + sh[threadIdx.x] - v;   // exclusive
    carry += sh[1023];
    __syncthreads();
  }
  if (threadIdx.x == 0) seg_off[S] = carry;            // == E
}

// Pass 3: write exclusive offsets for all segments
__global__ void scan_write(const unsigned* __restrict__ cnt,
                           const unsigned* __restrict__ blocksum,
                           unsigned* __restrict__ seg_off, int S) {
  __shared__ unsigned sh[256];
  int i = blockIdx.x * 256 + threadIdx.x;
  unsigned v = (i < S) ? cnt[i] : 0u;
  sh[threadIdx.x] = v;
  __syncthreads();
  for (int off = 1; off < 256; off <<= 1) {
    unsigned t = (threadIdx.x >= (unsigned)off) ? sh[threadIdx.x - off] : 0u;
    __syncthreads();
    sh[threadIdx.x] += t;
    __syncthreads();
  }
  if (i < S) seg_off[i] = blocksum[blockIdx.x] + sh[threadIdx.x] - v;
}

// ------------------------------------- bucket edge sources per (dst,rel)
__global__ void place_kernel(const long long* __restrict__ ei,
                             const long long* __restrict__ et,
                             const unsigned* __restrict__ seg_off,
                             unsigned* __restrict__ cursor,
                             unsigned* __restrict__ edge_src, int E) {
  int e = blockIdx.x * blockDim.x + threadIdx.x;
  if (e >= E) return;
  int src = (int)ei[e];
  int dst = (int)ei[(size_t)E + e];
  int r   = (int)et[e];
  unsigned seg = (unsigned)dst * NREL + (unsigned)r;
  unsigned pos = seg_off[seg] + atomicAdd(&cursor[seg], 1u);
  edge_src[pos] = (unsigned)src;
}

// -------------------------------------------------------- WMMA fp32 GEMM
// Block = 256 threads (8 waves); rows [n0,n0+16). A tile staged ONCE in LDS
// (+4 pad -> 16 rows on 16 distinct banks) and reused for all 4 B matrices
// (trans_0..2 = h@W[r], rootterm = h@root). Wave w owns column tile w; each
// A-fragment feeds 4 back-to-back v_wmma_f32_16x16x4_f32.
__global__ void rgcn_gemm(const float* __restrict__ h,
                          const float* __restrict__ W,
                          const float* __restrict__ root,
                          float* __restrict__ trans,
                          float* __restrict__ rootterm,
                          int N, int K) {
  extern __shared__ float ldsA[];          // 16 x (K+4)
  const int ldK = K + 4;
  const int n0  = blockIdx.x * 16;
  const int tid = threadIdx.x;

  for (int i = tid; i < 16 * K; i += blockDim.x) {
    int rr = i / K, cc = i % K;
    int row = n0 + rr;
    ldsA[rr * ldK + cc] = (row < N) ? h[(size_t)row * K + cc] : 0.0f;
  }
  __syncthreads();

  const int lane = tid & 31;
  const int wave = tid >> 5;
  const int c0   = wave * 16;
  const int arow = lane & 15;
  const int kk   = (lane >> 4) * 2;

  const float* B0 = W;
  const float* B1 = W + (size_t)K * HID;
  const float* B2 = W + (size_t)2 * K * HID;
  const float* B3 = root;

  v8f acc0 = {}, acc1 = {}, acc2 = {}, acc3 = {};
  for (int k = 0; k < K; k += 4) {
    v2f a, b;
    a.x = ldsA[arow * ldK + k + kk];
    a.y = ldsA[arow * ldK + k + kk + 1];
    const size_t o0 = (size_t)(k + kk)     * HID + c0 + arow;
    const size_t o1 = (size_t)(k + kk + 1) * HID + c0 + arow;
    b.x = B0[o0]; b.y = B0[o1];
    acc0 = __builtin_amdgcn_wmma_f32_16x16x4_f32(false, a, false, b, (short)0, acc0, false, false);
    b.x = B1[o0]; b.y = B1[o1];
    acc1 = __builtin_amdgcn_wmma_f32_16x16x4_f32(false, a, false, b, (short)0, acc1, false, false);
    b.x = B2[o0]; b.y = B2[o1];
    acc2 = __builtin_amdgcn_wmma_f32_16x16x4_f32(false, a, false, b, (short)0, acc2, false, false);
    b.x = B3[o0]; b.y = B3[o1];
    acc3 = __builtin_amdgcn_wmma_f32_16x16x4_f32(false, a, false, b, (short)0, acc3, false, false);
  }

  const int rbase = (lane >> 4) * 8;
  const int col   = c0 + (lane & 15);
  float* O0 = trans;
  float* O1 = trans + (size_t)N * HID;
  float* O2 = trans + (size_t)2 * N * HID;
  float* O3 = rootterm;
#pragma unroll
  for (int i = 0; i < 8; ++i) {
    int row = n0 + rbase + i;
    if (row < N) {
      O0[(size_t)row * HID + col] = acc0[i];
      O1[(size_t)row * HID + col] = acc1[i];
      O2[(size_t)row * HID + col] = acc2[i];
      O3[(size_t)row * HID + col] = acc3[i];
    }
  }
}

// -------------------- atomic-free gather: mean per (dst,rel), sum over rel
__device__ __forceinline__ float4 gather_node(int n,
                                              const unsigned* __restrict__ seg_off,
                                              const unsigned* __restrict__ edge_src,
                                              const float* __restrict__ trans,
                                              int N, int c0) {
  float4 acc = make_float4(0.f, 0.f, 0.f, 0.f);
#pragma unroll
  for (int r = 0; r < NREL; ++r) {
    const unsigned u0 = seg_off[n * NREL + r];
    const unsigned u1 = seg_off[n * NREL + r + 1];
    const float* base = trans + (size_t)r * N * HID;
    float4 p = make_float4(0.f, 0.f, 0.f, 0.f);
    unsigned i = u0;
    for (; i + 2 <= u1; i += 2) {          // 2 rows in flight
      unsigned s0 = edge_src[i], s1 = edge_src[i + 1];
      const float4 a = *(const float4*)(base + (size_t)s0 * HID + c0);
      const float4 b = *(const float4*)(base + (size_t)s1 * HID + c0);
      p.x += a.x + b.x; p.y += a.y + b.y; p.z += a.z + b.z; p.w += a.w + b.w;
    }
    if (i < u1) {
      unsigned s0 = edge_src[i];
      const float4 a = *(const float4*)(base + (size_t)s0 * HID + c0);
      p.x += a.x; p.y += a.y; p.z += a.z; p.w += a.w;
    }
    if (u1 > u0) {
      const float sc = 1.0f / (float)(u1 - u0);
      acc.x += p.x * sc; acc.y += p.y * sc; acc.z += p.z * sc; acc.w += p.w * sc;
    }
  }
  return acc;
}

// ------- layer 0: gather + root + bias -> LayerNorm -> LeakyReLU(0.2) -> h1
__global__ void gather_post0(const unsigned* __restrict__ seg_off,
                             const unsigned* __restrict__ edge_src,
                             const float* __restrict__ trans,
                             const float* __restrict__ rt,
                             const float* __restrict__ bias,
                             const float* __restrict__ gamma,
                             const float* __restrict__ beta,
                             float* __restrict__ out, int N) {
  const int wave = threadIdx.x >> 5, lane = threadIdx.x & 31;
  const int n = blockIdx.x * (blockDim.x >> 5) + wave;
  if (n >= N) return;
  const int c0 = lane * 4;

  float4 a = gather_node(n, seg_off, edge_src, trans, N, c0);
  const float4 r = *(const float4*)(rt + (size_t)n * HID + c0);
  const float4 b = *(const float4*)(bias + c0);
  float v[4] = {a.x + r.x + b.x, a.y + r.y + b.y, a.z + r.z + b.z, a.w + r.w + b.w};

  float mean = wave_sum32(v[0] + v[1] + v[2] + v[3]) * (1.0f / HID);
  float d[4], sq = 0.0f;
#pragma unroll
  for (int j = 0; j < 4; ++j) { d[j] = v[j] - mean; sq += d[j] * d[j]; }
  float rstd = rsqrtf(wave_sum32(sq) * (1.0f / HID) + 1e-5f);

  const float4 gm = *(const float4*)(gamma + c0);
  const float4 bt = *(const float4*)(beta  + c0);
  float y[4] = {d[0] * rstd * gm.x + bt.x, d[1] * rstd * gm.y + bt.y,
                d[2] * rstd * gm.z + bt.z, d[3] * rstd * gm.w + bt.w};
#pragma unroll
  for (int j = 0; j < 4; ++j) y[j] = (y[j] > 0.0f) ? y[j] : 0.2f * y[j];

  *(float4*)(out + (size_t)n * HID + c0) = make_float4(y[0], y[1], y[2], y[3]);
}

// ------- layer 1: gather + root + bias -> LN(g1,be1) -> LN(gout,bout) -> out
__global__ void gather_post1(const unsigned* __restrict__ seg_off,
                             const unsigned* __restrict__ edge_src,
                             const float* __restrict__ trans,
                             const float* __restrict__ rt,
                             const float* __restrict__ bias,
                             const float* __restrict__ gamma,
                             const float* __restrict__ beta,
                             const float* __restrict__ gamma2,
                             const float* __restrict__ beta2,
                             float* __restrict__ out, int N) {
  const int wave = threadIdx.x >> 5, lane = threadIdx.x & 31;
  const int n = blockIdx.x * (blockDim.x >> 5) + wave;
  if (n >= N) return;
  const int c0 = lane * 4;

  float4 a = gather_node(n, seg_off, edge_src, trans, N, c0);
  const float4 r = *(const float4*)(rt + (size_t)n * HID + c0);
  const float4 b = *(const float4*)(bias + c0);
  float v[4] = {a.x + r.x + b.x, a.y + r.y + b.y, a.z + r.z + b.z, a.w + r.w + b.w};

  // LN 1
  float mean = wave_sum32(v[0] + v[1] + v[2] + v[3]) * (1.0f / HID);
  float d[4], sq = 0.0f;
#pragma unroll
  for (int j = 0; j < 4; ++j) { d[j] = v[j] - mean; sq += d[j] * d[j]; }
  float rstd = rsqrtf(wave_sum32(sq) * (1.0f / HID) + 1e-5f);
  const float4 gm = *(const float4*)(gamma + c0);
  const float4 bt = *(const float4*)(beta  + c0);
  float y[4] = {d[0] * rstd * gm.x + bt.x, d[1] * rstd * gm.y + bt.y,
                d[2] * rstd * gm.z + bt.z, d[3] * rstd * gm.w + bt.w};

  // LN 2
  float mean2 = wave_sum32(y[0] + y[1] + y[2] + y[3]) * (1.0f / HID);
  float e[4], sq2 = 0.0f;
#pragma unroll
  for (int j = 0; j < 4; ++j) { e[j] = y[j] - mean2; sq2 += e[j] * e[j]; }
  float rstd2 = rsqrtf(wave_sum32(sq2) * (1.0f / HID) + 1e-5f);
  const float4 gm2 = *(const float4*)(gamma2 + c0);
  const float4 bt2 = *(const float4*)(beta2  + c0);
  float z[4] = {e[0] * rstd2 * gm2.x + bt2.x, e[1] * rstd2 * gm2.y + bt2.y,
                e[2] * rstd2 * gm2.z + bt2.z, e[3] * rstd2 * gm2.w + bt2.w};

  *(float4*)(out + (size_t)n * HID + c0) = make_float4(z[0], z[1], z[2], z[3]);
}

// --------------------------------------------------------------- launcher
extern "C" void kernel_launch(void* const* d_in, const int* in_sizes, int n_in,
                              void* d_out, int out_size, void* d_ws, size_t ws_size,
                              hipStream_t stream) {
  const float*     x     = (const float*)d_in[0];
  const long long* ei    = (const long long*)d_in[1];   // int64 per reference
  const long long* et    = (const long long*)d_in[2];   // int64 per reference
  const float*     W0    = (const float*)d_in[3];
  const float*     root0 = (const float*)d_in[4];
  const float*     b0    = (const float*)d_in[5];
  const float*     g0    = (const float*)d_in[6];
  const float*     be0   = (const float*)d_in[7];
  const float*     W1    = (const float*)d_in[8];
  const float*     root1 = (const float*)d_in[9];
  const float*     b1    = (const float*)d_in[10];
  const float*     g1    = (const float*)d_in[11];
  const float*     be1   = (const float*)d_in[12];
  const float*     gout  = (const float*)d_in[13];
  const float*     bout  = (const float*)d_in[14];

  const int N  = in_sizes[0] / IN_DIM;
  const int E  = in_sizes[2];
  const int S  = N * NREL;                 // segment count
  const int nb = (S + 255) / 256;          // scan blocks

  // workspace carve-out (256B aligned): ~265 MB total
  char* ws = (char*)d_ws;
  size_t off = 0;
  auto take = [&](size_t bytes) -> char* {
    char* p = ws + off;
    off = (off + bytes + 255) & ~(size_t)255;
    return p;
  };
  float*    trans    = (float*)take((size_t)NREL * N * HID * sizeof(float));
  float*    rootterm = (float*)take((size_t)N * HID * sizeof(float));
  float*    h1       = (float*)take((size_t)N * HID * sizeof(float));
  unsigned* cnt      = (unsigned*)take((size_t)S * sizeof(unsigned));
  unsigned* seg_off  = (unsigned*)take((size_t)(S + 1) * sizeof(unsigned));
  unsigned* cursor   = (unsigned*)take((size_t)S * sizeof(unsigned));
  unsigned* edge_src = (unsigned*)take((size_t)E * sizeof(unsigned));
  unsigned* blocksum = (unsigned*)take((size_t)nb * sizeof(unsigned));

  // ---- one-time CSR build (shared by both layers)
  hipMemsetAsync(cnt, 0, (size_t)S * sizeof(unsigned), stream);
  count_kernel<<<(E + 255) / 256, 256, 0, stream>>>(ei, et, cnt, E);
  scan_partials<<<nb, 256, 0, stream>>>(cnt, blocksum, S);
  scan_blocksums<<<1, 1024, 0, stream>>>(blocksum, nb, seg_off, S);
  scan_write<<<nb, 256, 0, stream>>>(cnt, blocksum, seg_off, S);
  hipMemsetAsync(cursor, 0, (size_t)S * sizeof(unsigned), stream);
  place_kernel<<<(E + 255) / 256, 256, 0, stream>>>(ei, et, seg_off, cursor, edge_src, E);

  const int gemm_blocks = (N + 15) / 16;
  const int node_blocks = (N + 7) / 8;     // 8 waves/block, wave per node

  // ---- layer 0 (K = 256)
  rgcn_gemm<<<gemm_blocks, 256, 16 * (IN_DIM + 4) * sizeof(float), stream>>>(
      x, W0, root0, trans, rootterm, N, IN_DIM);
  gather_post0<<<node_blocks, 256, 0, stream>>>(seg_off, edge_src, trans, rootterm,
                                                b0, g0, be0, h1, N);

  // ---- layer 1 (K = 128)
  rgcn_gemm<<<gemm_blocks, 256, 16 * (HID + 4) * sizeof(float), stream>>>(
      h1, W1, root1, trans, rootterm, N, HID);
  gather_post1<<<node_blocks, 256, 0, stream>>>(seg_off, edge_src, trans, rootterm,
                                                b1, g1, be1, gout, bout,
                                                (float*)d_out, N);
}